// FastRenderer_32186484916658
// MI455X (gfx1250) — compile-verified
//
#include <hip/hip_runtime.h>
#include <math.h>

#define RES          128
#define SDF_THR      5e-05f
#define NSTEPS       100
#define NSECANT      8
#define TRACE_ITERS  10

// ---------------------------------------------------------------------------
// Trilinear SDF query, exact port of trilinear_sdf().
// vol is [R,R,R] row-major (z contiguous): 8 corners = 4 adjacent float pairs,
// which the compiler merges into 4x global_load_b64.
// ---------------------------------------------------------------------------
__device__ __forceinline__ float tri_sdf(const float* __restrict__ vol,
                                         float px, float py, float pz) {
    float ux = (px + 1.0f) * 0.5f * 127.0f;
    float uy = (py + 1.0f) * 0.5f * 127.0f;
    float uz = (pz + 1.0f) * 0.5f * 127.0f;
    ux = fminf(fmaxf(ux, 0.0f), 127.0f);
    uy = fminf(fmaxf(uy, 0.0f), 127.0f);
    uz = fminf(fmaxf(uz, 0.0f), 127.0f);
    int ix = (int)floorf(ux); ix = ix < 0 ? 0 : (ix > 126 ? 126 : ix);
    int iy = (int)floorf(uy); iy = iy < 0 ? 0 : (iy > 126 ? 126 : iy);
    int iz = (int)floorf(uz); iz = iz < 0 ? 0 : (iz > 126 ? 126 : iz);
    float fx = ux - (float)ix;
    float fy = uy - (float)iy;
    float fz = uz - (float)iz;
    const float* b = vol + ((ix * RES + iy) * RES + iz);
    float v000 = b[0];
    float v001 = b[1];
    float v010 = b[RES];
    float v011 = b[RES + 1];
    float v100 = b[RES * RES];
    float v101 = b[RES * RES + 1];
    float v110 = b[RES * RES + RES];
    float v111 = b[RES * RES + RES + 1];
    float gx  = 1.0f - fx;
    float c00 = v000 * gx + v100 * fx;
    float c10 = v010 * gx + v110 * fx;
    float c01 = v001 * gx + v101 * fx;
    float c11 = v011 * gx + v111 * fx;
    float gy  = 1.0f - fy;
    float c0  = c00 * gy + c10 * fy;
    float c1  = c01 * gy + c11 * fy;
    return c0 * (1.0f - fz) + c1 * fz;
}

// ---------------------------------------------------------------------------
// Sphere tracing (exact port). Returns curr_s, acc_s, acc_e, unfin_s.
// ---------------------------------------------------------------------------
__device__ __forceinline__ void trace_ray(const float* __restrict__ vol,
                                          float ox, float oy, float oz,
                                          float rdx, float rdy, float rdz,
                                          float tn, float tf,
                                          float& sx, float& sy, float& sz,
                                          float& acc_s, float& acc_e,
                                          bool& unfin_s_out) {
    bool unfin_s = tn < tf;
    bool unfin_e = unfin_s;
    acc_s = tn; acc_e = tf;
    sx = ox + rdx*tn; sy = oy + rdy*tn; sz = oz + rdz*tn;
    float ex = ox + rdx*tf, ey = oy + rdy*tf, ez = oz + rdz*tf;
    float next_s = unfin_s ? tri_sdf(vol, sx, sy, sz) : 0.0f;
    float next_e = unfin_e ? tri_sdf(vol, ex, ey, ez) : 0.0f;

    #pragma unroll 1
    for (int it = 0; it < TRACE_ITERS; ++it) {
        float cs = unfin_s ? next_s : 0.0f;
        cs = (cs <= SDF_THR) ? 0.0f : cs;
        float ce = unfin_e ? next_e : 0.0f;
        ce = (ce <= SDF_THR) ? 0.0f : ce;
        unfin_s = unfin_s && (cs > SDF_THR);
        unfin_e = unfin_e && (ce > SDF_THR);
        acc_s += cs;
        acc_e -= ce;
        sx = ox + acc_s*rdx; sy = oy + acc_s*rdy; sz = oz + acc_s*rdz;
        ex = ox + acc_e*rdx; ey = oy + acc_e*rdy; ez = oz + acc_e*rdz;
        // two independent gather chains (s & e) -> MLP within one lane
        next_s = unfin_s ? tri_sdf(vol, sx, sy, sz) : 0.0f;
        next_e = unfin_e ? tri_sdf(vol, ex, ey, ez) : 0.0f;
        // line search: LINE_STEP_ITERS==1, step = 0.5
        if (next_s < 0.0f) {
            acc_s -= 0.5f * cs;
            sx = ox + acc_s*rdx; sy = oy + acc_s*rdy; sz = oz + acc_s*rdz;
            next_s = tri_sdf(vol, sx, sy, sz);
        }
        if (next_e < 0.0f) {
            acc_e += 0.5f * ce;
            ex = ox + acc_e*rdx; ey = oy + acc_e*rdy; ez = oz + acc_e*rdz;
            next_e = tri_sdf(vol, ex, ey, ez);
        }
        bool alive = acc_s < acc_e;
        unfin_s = unfin_s && alive;
        unfin_e = unfin_e && alive;
    }
    {   // final mask update (only unfin_s used downstream)
        float cs = unfin_s ? next_s : 0.0f;
        cs = (cs <= SDF_THR) ? 0.0f : cs;
        unfin_s = unfin_s && (cs > SDF_THR);
    }
    unfin_s_out = unfin_s;
}

// ---------------------------------------------------------------------------
// 100-step ray sampler + 8-step secant (exact port of ray_sampler for a
// sampler_mask==true row). Outputs sampler point / dist / net_surface mask.
// ---------------------------------------------------------------------------
__device__ __forceinline__ void run_sampler(const float* __restrict__ vol,
                                            float ox, float oy, float oz,
                                            float rdx, float rdy, float rdz,
                                            float zn, float zf,
                                            float& px_o, float& py_o, float& pz_o,
                                            float& pt_o, float& pm_o) {
    float zr = zf - zn;

    // Warm L0 with the mid-interval sample line (gfx1250 global_prefetch_b8).
    {
        float zm  = zn + 0.5f * zr;
        float pmx = ox + zm*rdx, pmy = oy + zm*rdy, pmz = oz + zm*rdz;
        int jx = (int)((pmx + 1.0f) * 63.5f); jx = jx < 0 ? 0 : (jx > 126 ? 126 : jx);
        int jy = (int)((pmy + 1.0f) * 63.5f); jy = jy < 0 ? 0 : (jy > 126 ? 126 : jy);
        int jz = (int)((pmz + 1.0f) * 63.5f); jz = jz < 0 ? 0 : (jz > 126 ? 126 : jz);
        __builtin_prefetch(vol + ((jx * RES + jy) * RES + jz), 0, 1);
    }

    // argmin of sign(sdf)*arange(100,0,-1) (first occurrence) and argmin(sdf)
    float best_tmp = INFINITY;
    int   ind      = 0;
    float sdf_ind  = 0.0f, z_ind = zn;
    float sdf_low0 = 0.0f, z_low0 = 0.0f;   // values at ind-1
    float best_sdf = INFINITY;
    float z_out    = zn;
    float prev_sdf = 0.0f, prev_z = 0.0f;

    #pragma unroll 4   // ~16 b64 gathers in flight per lane
    for (int k = 0; k < NSTEPS; ++k) {
        float t  = (float)k / 99.0f;
        float z  = zn + t * zr;
        float px = ox + z*rdx, py = oy + z*rdy, pz = oz + z*rdz;
        float sv = tri_sdf(vol, px, py, pz);
        float sgn = (sv > 0.0f) ? 1.0f : ((sv < 0.0f) ? -1.0f : 0.0f);
        float tmp = sgn * (float)(NSTEPS - k);
        if (tmp < best_tmp) {
            best_tmp = tmp; ind = k;
            sdf_ind = sv; z_ind = z;
            sdf_low0 = prev_sdf; z_low0 = prev_z;
        }
        if (sv < best_sdf) { best_sdf = sv; z_out = z; }
        prev_sdf = sv; prev_z = z;
    }
    // JAX negative-index wrap: ind-1 == -1 gathers element 99
    if (ind == 0) { sdf_low0 = prev_sdf; z_low0 = prev_z; }

    bool  net_surface = sdf_ind < 0.0f;
    float sp_t = net_surface ? z_ind : z_out;

    if (net_surface) {   // secant refinement (secant_rows == true here)
        float z_low = z_low0,  sdf_low  = sdf_low0;
        float z_high = z_ind,  sdf_high = sdf_ind;
        float denom = sdf_high - sdf_low;
        float safe  = (fabsf(denom) > 1e-12f) ? denom : 1.0f;
        float z_pred = -sdf_low * (z_high - z_low) / safe + z_low;
        #pragma unroll 1
        for (int s = 0; s < NSECANT; ++s) {
            float px = ox + z_pred*rdx, py = oy + z_pred*rdy, pz = oz + z_pred*rdz;
            float sm = tri_sdf(vol, px, py, pz);
            if (sm > 0.0f) { z_low  = z_pred; sdf_low  = sm; }
            if (sm < 0.0f) { z_high = z_pred; sdf_high = sm; }
            denom = sdf_high - sdf_low;
            safe  = (fabsf(denom) > 1e-12f) ? denom : 1.0f;
            z_pred = -sdf_low * (z_high - z_low) / safe + z_low;
        }
        sp_t = z_pred;
    }
    px_o = ox + sp_t*rdx; py_o = oy + sp_t*rdy; pz_o = oz + sp_t*rdz;
    pt_o = sp_t;
    pm_o = net_surface ? 1.0f : 0.0f;   // s_obj (sampler_mask is true here)
}

// ---------------------------------------------------------------------------
// Kernel 0: zero the compaction counter (graph-capture safe).
// ---------------------------------------------------------------------------
__global__ void zero_counter_kernel(unsigned int* c) { *c = 0u; }

// ---------------------------------------------------------------------------
// Kernel 1: sphere tracing + default outputs + block-aggregated compaction of
// unfinished rays into a queue. out[4N+i] (= acc_s) doubles as zn for phase 2.
// ---------------------------------------------------------------------------
__global__ void __launch_bounds__(256)
trace_kernel(const float* __restrict__ rays_o,
             const float* __restrict__ rays_d,
             const float* __restrict__ nearv,
             const float* __restrict__ farv,
             const float* __restrict__ vol,
             float* __restrict__ out, int N,
             unsigned int* __restrict__ counter,
             int* __restrict__ queue,
             float* __restrict__ acc_e_ws) {
    __shared__ unsigned int s_cnt, s_base;
    int i = blockIdx.x * blockDim.x + threadIdx.x;
    bool active = i < N;
    bool sampler_mask = false;

    if (threadIdx.x == 0) s_cnt = 0u;
    __syncthreads();

    if (active) {
        // ray data is streamed exactly once -> non-temporal (keep volume in L2)
        float ox  = __builtin_nontemporal_load(&rays_o[3*i+0]);
        float oy  = __builtin_nontemporal_load(&rays_o[3*i+1]);
        float oz  = __builtin_nontemporal_load(&rays_o[3*i+2]);
        float rdx = __builtin_nontemporal_load(&rays_d[3*i+0]);
        float rdy = __builtin_nontemporal_load(&rays_d[3*i+1]);
        float rdz = __builtin_nontemporal_load(&rays_d[3*i+2]);
        float tn  = __builtin_nontemporal_load(&nearv[i]);
        float tf  = __builtin_nontemporal_load(&farv[i]);

        float sx, sy, sz, acc_s, acc_e;
        bool unfin_s;
        trace_ray(vol, ox, oy, oz, rdx, rdy, rdz, tn, tf,
                  sx, sy, sz, acc_s, acc_e, unfin_s);

        sampler_mask = unfin_s;
        bool net_obj = acc_s < acc_e;

        // defaults (phase 2 overwrites sampler rays; out_t carries zn=acc_s)
        out[3*i+0]   = sx;
        out[3*i+1]   = sy;
        out[3*i+2]   = sz;
        out[3*N + i] = net_obj ? 1.0f : 0.0f;
        out[4*N + i] = acc_s;
        if (sampler_mask) acc_e_ws[i] = acc_e;
    }

    // block-aggregated queue append: 1 LDS atomic/lane, 1 global atomic/block
    unsigned int local = 0u;
    if (sampler_mask) local = atomicAdd(&s_cnt, 1u);
    __syncthreads();
    if (threadIdx.x == 0 && s_cnt != 0u) s_base = atomicAdd(counter, s_cnt);
    __syncthreads();
    if (sampler_mask) queue[s_base + local] = i;
}

// ---------------------------------------------------------------------------
// Kernel 2: compacted sampler + secant. Launched at worst-case grid; threads
// beyond the device-side count exit immediately (grid must be static for
// graph capture). Full-lane waves in the gather-heavy phase.
// ---------------------------------------------------------------------------
__global__ void __launch_bounds__(256)
sampler_kernel(const float* __restrict__ rays_o,
               const float* __restrict__ rays_d,
               const float* __restrict__ vol,
               const float* __restrict__ acc_e_ws,
               const int* __restrict__ queue,
               const unsigned int* __restrict__ counter,
               float* __restrict__ out, int N) {
    unsigned int t = blockIdx.x * blockDim.x + threadIdx.x;
    if (t >= *counter) return;
    int i = queue[t];

    float ox  = rays_o[3*i+0], oy  = rays_o[3*i+1], oz  = rays_o[3*i+2];
    float rdx = rays_d[3*i+0], rdy = rays_d[3*i+1], rdz = rays_d[3*i+2];
    float zn  = out[4*N + i];        // acc_s written by trace_kernel
    float zf  = acc_e_ws[i];

    float px, py, pz, pt, pm;
    run_sampler(vol, ox, oy, oz, rdx, rdy, rdz, zn, zf, px, py, pz, pt, pm);

    out[3*i+0]   = px;
    out[3*i+1]   = py;
    out[3*i+2]   = pz;
    out[3*N + i] = pm;
    out[4*N + i] = pt;
}

// ---------------------------------------------------------------------------
// Fallback: monolithic kernel (used only if workspace is too small).
// ---------------------------------------------------------------------------
__global__ void __launch_bounds__(256)
mono_kernel(const float* __restrict__ rays_o,
            const float* __restrict__ rays_d,
            const float* __restrict__ nearv,
            const float* __restrict__ farv,
            const float* __restrict__ vol,
            float* __restrict__ out, int N) {
    int i = blockIdx.x * blockDim.x + threadIdx.x;
    if (i >= N) return;
    float ox = rays_o[3*i+0], oy = rays_o[3*i+1], oz = rays_o[3*i+2];
    float rdx = rays_d[3*i+0], rdy = rays_d[3*i+1], rdz = rays_d[3*i+2];
    float tn = nearv[i], tf = farv[i];

    float sx, sy, sz, acc_s, acc_e;
    bool unfin_s;
    trace_ray(vol, ox, oy, oz, rdx, rdy, rdz, tn, tf,
              sx, sy, sz, acc_s, acc_e, unfin_s);

    bool net_obj = acc_s < acc_e;
    float out_x = sx, out_y = sy, out_z = sz;
    float out_t = acc_s;
    float out_m = net_obj ? 1.0f : 0.0f;

    if (unfin_s) {
        run_sampler(vol, ox, oy, oz, rdx, rdy, rdz, acc_s, acc_e,
                    out_x, out_y, out_z, out_t, out_m);
    }
    out[3*i+0]   = out_x;
    out[3*i+1]   = out_y;
    out[3*i+2]   = out_z;
    out[3*N + i] = out_m;
    out[4*N + i] = out_t;
}

extern "C" void kernel_launch(void* const* d_in, const int* in_sizes, int n_in,
                              void* d_out, int out_size, void* d_ws, size_t ws_size,
                              hipStream_t stream) {
    (void)n_in; (void)out_size;
    const float* rays_o = (const float*)d_in[0];
    const float* rays_d = (const float*)d_in[1];
    const float* nearv  = (const float*)d_in[2];
    const float* farv   = (const float*)d_in[3];
    const float* vol    = (const float*)d_in[4];
    float* out = (float*)d_out;
    int N = in_sizes[2];           // near has N elements (rays_o is 3N)

    dim3 block(256);               // 8 wave32 waves per workgroup
    dim3 grid((N + 255) / 256);

    size_t need = 256 + (size_t)N * 4 /*queue*/ + (size_t)N * 4 /*acc_e*/;
    if (d_ws != nullptr && ws_size >= need) {
        unsigned int* counter  = (unsigned int*)d_ws;
        int*          queue    = (int*)((char*)d_ws + 256);
        float*        acc_e_ws = (float*)((char*)d_ws + 256 + (size_t)N * 4);

        hipLaunchKernelGGL(zero_counter_kernel, dim3(1), dim3(1), 0, stream, counter);
        hipLaunchKernelGGL(trace_kernel, grid, block, 0, stream,
                           rays_o, rays_d, nearv, farv, vol, out, N,
                           counter, queue, acc_e_ws);
        hipLaunchKernelGGL(sampler_kernel, grid, block, 0, stream,
                           rays_o, rays_d, vol, acc_e_ws, queue, counter, out, N);
    } else {
        hipLaunchKernelGGL(mono_kernel, grid, block, 0, stream,
                           rays_o, rays_d, nearv, farv, vol, out, N);
    }
}